// MyGCN_89455578841530
// MI455X (gfx1250) — compile-verified
//
#include <hip/hip_runtime.h>
#include <cmath>

#define N_NODES 100000
#define DIM     256
#define NEDGE   1600000
#define SSENT   512
#define EMAXN   8
#define HDIM    256
#define EPSV    1e-6f

typedef float v2f __attribute__((ext_vector_type(2)));
typedef float v8f __attribute__((ext_vector_type(8)));

#define WMMA_F32(a, b, c) \
  __builtin_amdgcn_wmma_f32_16x16x4_f32(false, (a), false, (b), (short)0, (c), false, false)

// ---------------------------------------------------------------- utilities
__global__ void zero_f32(float* __restrict__ p, long n) {
  long i = (long)blockIdx.x * blockDim.x + threadIdx.x;
  if (i < n) p[i] = 0.f;
}

__global__ void relu_f32(float* __restrict__ p, long n) {
  long i = (long)blockIdx.x * blockDim.x + threadIdx.x;
  if (i < n) { float v = p[i]; p[i] = v > 0.f ? v : 0.f; }
}

// ---------------------------------------------------------------- graph norm
__global__ void deg_kernel(const int* __restrict__ dst, float* __restrict__ deg, int E) {
  int e = blockIdx.x * blockDim.x + threadIdx.x;
  if (e < E) atomicAdd(&deg[dst[e]], 1.0f);
}

__global__ void dis_kernel(float* __restrict__ deg, int n) {
  int i = blockIdx.x * blockDim.x + threadIdx.x;
  if (i < n) { float v = deg[i]; deg[i] = (v > 0.f) ? rsqrtf(v) : 0.f; }
}

__global__ void norm_kernel(const int* __restrict__ src, const int* __restrict__ dst,
                            const float* __restrict__ dis, float* __restrict__ nrm, int E) {
  int e = blockIdx.x * blockDim.x + threadIdx.x;
  if (e < E) nrm[e] = dis[dst[e]] * dis[src[e]];
}

// ---------------------------------------------------------------- WMMA GEMMs
// C[M,256] = A[M,256] @ B[256,256] (B row-major, B[k][n]).
// One wave computes a 16x64 strip: 4 accumulators share one A fragment per K-step.
// 25000 wave-strips (6250 m-tiles x 4 col-slabs), 8 waves/block -> 3125 blocks.
__global__ void __launch_bounds__(256)
gemm_conv_wmma(const float* __restrict__ A, const float* __restrict__ B,
               float* __restrict__ C) {
  int wave = threadIdx.x >> 5;
  int lane = threadIdx.x & 31;
  int unit = blockIdx.x * 8 + wave;
  int tm   = unit >> 2;                  // m-tile 0..6249
  int slab = unit & 3;                   // 64-column slab
  int half = lane >> 4, lm = lane & 15;
  const float* arow = A + (long)(tm * 16 + lm) * DIM;
  int col0 = slab * 64 + lm;
  v8f acc0 = {}, acc1 = {}, acc2 = {}, acc3 = {};
  for (int k0 = 0; k0 < DIM; k0 += 4) {
    int ka = k0 + 2 * half;
    v2f a = *(const v2f*)(arow + ka);                  // A[m][ka], A[m][ka+1]
    const float* b0 = B + (long)ka * DIM + col0;       // B[ka][col]
    const float* b1 = b0 + DIM;                        // B[ka+1][col]
    v2f b;
    b.x = b0[0];  b.y = b1[0];  acc0 = WMMA_F32(a, b, acc0);
    b.x = b0[16]; b.y = b1[16]; acc1 = WMMA_F32(a, b, acc1);
    b.x = b0[32]; b.y = b1[32]; acc2 = WMMA_F32(a, b, acc2);
    b.x = b0[48]; b.y = b1[48]; acc3 = WMMA_F32(a, b, acc3);
  }
  float* crow = C + (long)(tm * 16 + 8 * half) * DIM + col0;
#pragma unroll
  for (int g = 0; g < 8; ++g) {
    crow[(long)g * DIM + 0]  = acc0[g];
    crow[(long)g * DIM + 16] = acc1[g];
    crow[(long)g * DIM + 32] = acc2[g];
    crow[(long)g * DIM + 48] = acc3[g];
  }
}

// lin[4096,256] = relu( h[eidx[r]] @ sent_W^T + sent_b ).  B^T access -> row-wise float2.
__global__ void __launch_bounds__(256)
sent_lin_wmma(const float* __restrict__ h, const int* __restrict__ eidx,
              const float* __restrict__ sW, const float* __restrict__ sb,
              float* __restrict__ lin) {
  int wave = threadIdx.x >> 5;
  int lane = threadIdx.x & 31;
  int tile = blockIdx.x * 8 + wave;
  int tm = tile >> 4, tn = tile & 15;                  // 256 x 16 tiles
  int half = lane >> 4, lm = lane & 15;
  int node = eidx[tm * 16 + lm];
  const float* arow = h + (long)node * DIM;
  int col = tn * 16 + lm;
  const float* brow = sW + (long)col * DIM;            // transposed B
  v8f acc = {};
  for (int k0 = 0; k0 < DIM; k0 += 4) {
    int ka = k0 + 2 * half;
    v2f a = *(const v2f*)(arow + ka);
    v2f b = *(const v2f*)(brow + ka);
    acc = WMMA_F32(a, b, acc);
  }
  float bias = sb[col];
  int rbase = tm * 16 + 8 * half;
#pragma unroll
  for (int g = 0; g < 8; ++g) {
    float v = acc[g] + bias;
    lin[(long)(rbase + g) * DIM + col] = v > 0.f ? v : 0.f;
  }
}

// gatesIn[512,1024] = sent_feat[512,256] @ W_ih^T + b_ih + b_hh
__global__ void __launch_bounds__(256)
gates_in_wmma(const float* __restrict__ sf, const float* __restrict__ Wih,
              const float* __restrict__ bih, const float* __restrict__ bhh,
              float* __restrict__ gout) {
  int wave = threadIdx.x >> 5;
  int lane = threadIdx.x & 31;
  int tile = blockIdx.x * 8 + wave;
  int tm = tile >> 6, tn = tile & 63;                  // 32 x 64 tiles
  int half = lane >> 4, lm = lane & 15;
  const float* arow = sf + (long)(tm * 16 + lm) * DIM;
  int col = tn * 16 + lm;
  const float* brow = Wih + (long)col * DIM;           // transposed B
  v8f acc = {};
  for (int k0 = 0; k0 < DIM; k0 += 4) {
    int ka = k0 + 2 * half;
    v2f a = *(const v2f*)(arow + ka);
    v2f b = *(const v2f*)(brow + ka);
    acc = WMMA_F32(a, b, acc);
  }
  float bias = bih[col] + bhh[col];
  int rbase = tm * 16 + 8 * half;
#pragma unroll
  for (int g = 0; g < 8; ++g)
    gout[(long)(rbase + g) * (4 * HDIM) + col] = acc[g] + bias;
}

// ---------------------------------------------------------------- edge stage
__global__ void __launch_bounds__(256)
scatter_kernel(const float* __restrict__ tmp, const int* __restrict__ src,
               const int* __restrict__ dst, const float* __restrict__ nrm,
               float* __restrict__ h, int E) {
  int gid = blockIdx.x * 256 + threadIdx.x;
  int e = gid >> 6;                 // 64 threads per edge
  int q = gid & 63;                 // each thread: 4 consecutive floats
  if (e < E) {
    int s = src[e], d = dst[e];
    float w = nrm[e];
    const float4 v = *(const float4*)(tmp + (long)s * DIM + q * 4);
    float* p = h + (long)d * DIM + q * 4;
    atomicAdd(p + 0, w * v.x);
    atomicAdd(p + 1, w * v.y);
    atomicAdd(p + 2, w * v.z);
    atomicAdd(p + 3, w * v.w);
  }
}

// ---------------------------------------------------------------- sentences
__global__ void sent_reduce(const float* __restrict__ lin, const int* __restrict__ mask,
                            float* __restrict__ sf) {
  int i = blockIdx.x * blockDim.x + threadIdx.x;       // S*D
  if (i < SSENT * DIM) {
    int s = i / DIM, d = i % DIM;
    float sum = 0.f, cnt = 0.f;
#pragma unroll
    for (int j = 0; j < EMAXN; ++j) {
      float m = (float)mask[s * EMAXN + j];
      sum += m * lin[(long)(s * EMAXN + j) * DIM + d];
      cnt += m;
    }
    sf[i] = sum / (EPSV + cnt);
  }
}

// ---------------------------------------------------------------- LSTM
__global__ void transpose_whh(const float* __restrict__ W, float* __restrict__ WT) {
  int i = blockIdx.x * blockDim.x + threadIdx.x;       // k*1024 + g
  if (i < 4 * HDIM * HDIM) {
    int g = i % (4 * HDIM);
    int k = i / (4 * HDIM);
    WT[i] = W[(long)g * HDIM + k];
  }
}

__global__ void __launch_bounds__(1024)
lstm_kernel(const float* __restrict__ gatesIn, const float* __restrict__ WT,
            float* __restrict__ out) {
  __shared__ float h_s[HDIM];
  __shared__ float c_s[HDIM];
  __shared__ float gates[4 * HDIM];
  int tid = threadIdx.x;
  if (tid < HDIM) { h_s[tid] = 0.f; c_s[tid] = 0.f; }
  __syncthreads();
  for (int t = 0; t < SSENT; ++t) {
    float acc = gatesIn[(long)t * 4 * HDIM + tid];
#pragma unroll 4
    for (int k = 0; k < HDIM; ++k)
      acc += h_s[k] * WT[(long)k * 4 * HDIM + tid];    // coalesced: W_hh^T
    gates[tid] = acc;
    __syncthreads();
    if (tid < HDIM) {
      float ig = 1.f / (1.f + expf(-gates[tid]));
      float fg = 1.f / (1.f + expf(-gates[HDIM + tid]));
      float gg = tanhf(gates[2 * HDIM + tid]);
      float og = 1.f / (1.f + expf(-gates[3 * HDIM + tid]));
      float c = fg * c_s[tid] + ig * gg;
      c_s[tid] = c;
      float hn = og * tanhf(c);
      h_s[tid] = hn;
      out[(long)t * HDIM + tid] = hn;
    }
    __syncthreads();
  }
}

// ---------------------------------------------------------------- final head
__global__ void __launch_bounds__(512)
final_kernel(const float* __restrict__ lout, const float* __restrict__ nsp,
             const float* __restrict__ cls, const float* __restrict__ redW,
             const float* __restrict__ redb, const float* __restrict__ finW,
             const float* __restrict__ finb, float* __restrict__ out) {
  __shared__ float comb[512];
  int tid = threadIdx.x;
  if (tid < HDIM) {
    float acc = 0.f;
    for (int s = 0; s < SSENT - 1; ++s)
      acc += nsp[s] * (lout[(long)s * HDIM + tid] + lout[(long)(s + 1) * HDIM + tid]);
    comb[tid] = acc;
  } else {
    int j = tid - HDIM;
    float acc = redb[j];
    for (int k = 0; k < DIM; ++k) acc += cls[k] * redW[(long)j * DIM + k];
    comb[tid] = acc;
  }
  __syncthreads();
  if (tid == 0) {
    float o = finb[0];
    for (int k = 0; k < 512; ++k) o += comb[k] * finW[k];
    o = o > 0.f ? o : 0.f;
    out[0] = 1.f / (1.f + expf(-o));
  }
}

// ---------------------------------------------------------------- launcher
extern "C" void kernel_launch(void* const* d_in, const int* in_sizes, int n_in,
                              void* d_out, int out_size, void* d_ws, size_t ws_size,
                              hipStream_t stream) {
  const float* x     = (const float*)d_in[0];
  const int*   edge  = (const int*)d_in[1];
  const int*   srcI  = edge;
  const int*   dstI  = edge + NEDGE;
  const int*   eidx  = (const int*)d_in[2];
  const int*   emask = (const int*)d_in[3];
  const float* nsp   = (const float*)d_in[4];
  const float* cls   = (const float*)d_in[5];
  const float* convW = (const float*)d_in[6];
  const float* sentW = (const float*)d_in[7];
  const float* sentB = (const float*)d_in[8];
  const float* Wih   = (const float*)d_in[9];
  const float* Whh   = (const float*)d_in[10];
  const float* bih   = (const float*)d_in[11];
  const float* bhh   = (const float*)d_in[12];
  const float* redW  = (const float*)d_in[13];
  const float* redb  = (const float*)d_in[14];
  const float* finW  = (const float*)d_in[15];
  const float* finb  = (const float*)d_in[16];
  float* out = (float*)d_out;

  char* wp = (char*)d_ws;
  auto alloc = [&](size_t bytes) -> float* {
    float* p = (float*)wp;
    wp += (bytes + 255) & ~(size_t)255;
    return p;
  };
  float* h    = alloc((size_t)N_NODES * DIM * 4);        // node features
  float* tmp  = alloc((size_t)N_NODES * DIM * 4);        // h @ W
  float* deg  = alloc((size_t)N_NODES * 4);              // degree -> deg^-1/2
  float* nrm  = alloc((size_t)NEDGE * 4);                // per-edge weight
  float* lin  = alloc((size_t)SSENT * EMAXN * DIM * 4);  // entity linear
  float* sf   = alloc((size_t)SSENT * DIM * 4);          // sentence features
  float* gin  = alloc((size_t)SSENT * 4 * HDIM * 4);     // precomputed input gates
  float* WT   = alloc((size_t)4 * HDIM * HDIM * 4);      // W_hh transposed
  float* lout = alloc((size_t)SSENT * HDIM * 4);         // lstm outputs

  const long ND = (long)N_NODES * DIM;

  // degree normalization
  zero_f32<<<(N_NODES + 255) / 256, 256, 0, stream>>>(deg, N_NODES);
  deg_kernel<<<(NEDGE + 255) / 256, 256, 0, stream>>>(dstI, deg, NEDGE);
  dis_kernel<<<(N_NODES + 255) / 256, 256, 0, stream>>>(deg, N_NODES);
  norm_kernel<<<(NEDGE + 255) / 256, 256, 0, stream>>>(srcI, dstI, deg, nrm, NEDGE);

  const int convBlocks    = (N_NODES / 16) * 4 / 8;      // 25000 wave-strips -> 3125
  const int scatterBlocks = (NEDGE * 64) / 256;          // 400000
  const int ndBlocks      = (int)((ND + 255) / 256);

  // GCN layer 0: x -> tmp -> scatter into h -> relu
  gemm_conv_wmma<<<convBlocks, 256, 0, stream>>>(x, convW, tmp);
  zero_f32<<<ndBlocks, 256, 0, stream>>>(h, ND);
  scatter_kernel<<<scatterBlocks, 256, 0, stream>>>(tmp, srcI, dstI, nrm, h, NEDGE);
  relu_f32<<<ndBlocks, 256, 0, stream>>>(h, ND);

  // GCN layer 1: h -> tmp -> scatter into h -> relu
  gemm_conv_wmma<<<convBlocks, 256, 0, stream>>>(h, convW + DIM * DIM, tmp);
  zero_f32<<<ndBlocks, 256, 0, stream>>>(h, ND);
  scatter_kernel<<<scatterBlocks, 256, 0, stream>>>(tmp, srcI, dstI, nrm, h, NEDGE);
  relu_f32<<<ndBlocks, 256, 0, stream>>>(h, ND);

  // sentence features
  sent_lin_wmma<<<(SSENT * EMAXN / 16) * 16 / 8, 256, 0, stream>>>(h, eidx, sentW, sentB, lin);
  sent_reduce<<<(SSENT * DIM) / 256, 256, 0, stream>>>(lin, emask, sf);

  // LSTM
  gates_in_wmma<<<(SSENT / 16) * (4 * HDIM / 16) / 8, 256, 0, stream>>>(sf, Wih, bih, bhh, gin);
  transpose_whh<<<(4 * HDIM * HDIM) / 256, 256, 0, stream>>>(Whh, WT);
  lstm_kernel<<<1, 1024, 0, stream>>>(gin, WT, lout);

  // head
  final_kernel<<<1, 512, 0, stream>>>(lout, nsp, cls, redW, redb, finW, finb, out);
}